// SimGNN_74431783239843
// MI455X (gfx1250) — compile-verified
//
#include <hip/hip_runtime.h>
#include <hip/hip_bf16.h>

typedef float v2f __attribute__((ext_vector_type(2)));
typedef float v8f __attribute__((ext_vector_type(8)));

#define D0 64          // input feature dim
#define DMAX 128       // max layer width

// ---------------------------------------------------------------------------
// generic fill
__global__ void fill_kernel(float* __restrict__ p, size_t n, float v) {
    size_t i = (size_t)blockIdx.x * blockDim.x + threadIdx.x;
    size_t stride = (size_t)gridDim.x * blockDim.x;
    for (; i < n; i += stride) p[i] = v;
}

// degree count: deg[dst[e]] += 1
__global__ void count_deg_kernel(const int* __restrict__ ei, float* __restrict__ deg, int E) {
    int e = blockIdx.x * blockDim.x + threadIdx.x;
    if (e >= E) return;
    int d = ei[E + e];
    atomicAdd(&deg[d], 1.0f);
}

__global__ void rsqrt_kernel(float* __restrict__ p, int n) {
    int i = blockIdx.x * blockDim.x + threadIdx.x;
    if (i < n) p[i] = rsqrtf(p[i]);
}

// ---------------------------------------------------------------------------
// C[M x Nout] = act(A)[M x K] @ B[K x Nout], fp32 WMMA, LDS-staged A tile.
// One block = one 16-row M tile; wave w owns N-tile w (blockDim = 32*Nout/16).
// Fragment layouts per CDNA5 ISA 7.12.2:
//   A 16x4: lanes 0-15 hold (M=lane, K=k+0/k+1), lanes 16-31 K=k+2/k+3
//   B 4x16: lanes 0-15 hold (N=lane, K=k+0/k+1), lanes 16-31 K=k+2/k+3
//   C 16x16: VGPR r -> lanes 0-15 M=r, lanes 16-31 M=r+8; N=lane%16
__global__ __launch_bounds__(256)
void gemm_wmma_f32_lds(const float* __restrict__ A, const float* __restrict__ B,
                       float* __restrict__ C, int M, int K, int Nout,
                       int kshift, int relu_in) {
    __shared__ float As[16 * DMAX];      // 8 KB max
    const int tid  = threadIdx.x;
    const int lane = tid & 31;
    const int wave = tid >> 5;           // N-tile owned by this wave
    const int half = lane >> 4;
    const int l16  = lane & 15;
    const int m0   = blockIdx.x * 16;

    // stage A tile [16 x K] into LDS (float4 per thread), fused input ReLU
    const int vecs = (16 << kshift) >> 2;            // (16*K)/4
    for (int v = tid; v < vecs; v += blockDim.x) {
        int idx = v << 2;
        int r = idx >> kshift;
        int c = idx & (K - 1);
        int mr = m0 + r; if (mr >= M) mr = M - 1;    // clamped row
        float4 val = *(const float4*)(A + (size_t)mr * K + c);
        if (relu_in) {
            val.x = fmaxf(val.x, 0.0f); val.y = fmaxf(val.y, 0.0f);
            val.z = fmaxf(val.z, 0.0f); val.w = fmaxf(val.w, 0.0f);
        }
        *(float4*)(As + (r << kshift) + c) = val;
    }
    __syncthreads();

    const int n0 = wave * 16;
    const float* arow = As + (l16 << kshift) + 2 * half;
    const float* bcol = B + (size_t)(2 * half) * Nout + (n0 + l16);

    v8f acc = {};
    for (int k = 0; k < K; k += 4) {
        v2f a = *(const v2f*)(arow + k);             // ds_load_b64
        v2f b;
        b.x = bcol[(size_t)k * Nout];
        b.y = bcol[(size_t)(k + 1) * Nout];
        acc = __builtin_amdgcn_wmma_f32_16x16x4_f32(
            /*neg_a=*/false, a, /*neg_b=*/false, b,
            /*c_mod=*/(short)0, acc, /*reuse_a=*/false, /*reuse_b=*/false);
    }

    const int mbase = m0 + half * 8;
#pragma unroll
    for (int r = 0; r < 8; ++r) {
        int m = mbase + r;
        if (m < M) C[(size_t)m * Nout + n0 + l16] = acc[r];
    }
}

// ---------------------------------------------------------------------------
// initialize aggregation output with self-loop + bias (pure write):
//   out[v,c] = xw[v,c]*dinv[v]^2 + bias[c]
__global__ void init_out_kernel(const float* __restrict__ xw,
                                const float* __restrict__ dinv,
                                const float* __restrict__ bias,
                                float* __restrict__ out,
                                int N, int D, int dshift) {
    size_t i = (size_t)blockIdx.x * blockDim.x + threadIdx.x;
    size_t total = (size_t)N << dshift;
    if (i >= total) return;
    int row = (int)(i >> dshift);
    int col = (int)i & (D - 1);
    float dv = dinv[row];
    out[i] = xw[i] * dv * dv + bias[col];
}

// edge scatter: out[dst] += xw[src] * dinv[src]*dinv[dst]; 4 cols per thread,
// grid-stride with software prefetch of the next edge's source row.
__global__ void aggregate_kernel(const float* __restrict__ xw, float* __restrict__ out,
                                 const int* __restrict__ ei, const float* __restrict__ dinv,
                                 int E, int D, int gshift) {
    const int gmask = (1 << gshift) - 1;
    long long total = (long long)E << gshift;
    long long stride = (long long)gridDim.x * blockDim.x;
    for (long long t = (long long)blockIdx.x * blockDim.x + threadIdx.x;
         t < total; t += stride) {
        int e = (int)(t >> gshift);
        int g = (int)t & gmask;
        int s = ei[e];
        int d = ei[E + e];
        long long t2 = t + stride;
        if (t2 < total) {                            // prefetch next iteration's gather
            int s2 = ei[(int)(t2 >> gshift)];
            __builtin_prefetch(xw + ((size_t)s2 << (gshift + 2)) + (((int)t2 & gmask) << 2), 0, 0);
        }
        float w = dinv[s] * dinv[d];
        const float4 v = *(const float4*)(xw + (size_t)s * D + (g << 2));
        float* o = out + (size_t)d * D + (g << 2);
        atomicAdd(o + 0, v.x * w);
        atomicAdd(o + 1, v.y * w);
        atomicAdd(o + 2, v.z * w);
        atomicAdd(o + 3, v.w * w);
    }
}

// ---------------------------------------------------------------------------
// column sum of x[N x 64] into gsum[64]
__global__ void colsum_kernel(const float* __restrict__ x, float* __restrict__ gsum, int N) {
    int c   = threadIdx.x & 63;
    int sub = threadIdx.x >> 6;            // 0..3
    float acc = 0.0f;
    for (long long v = (long long)blockIdx.x * 4 + sub; v < N; v += (long long)gridDim.x * 4)
        acc += x[v * 64 + c];
    atomicAdd(&gsum[c], acc);
}

// gc = tanh((gsum/N) @ Wa), 1 block x 64 threads
__global__ void gc_kernel(const float* __restrict__ gsum, const float* __restrict__ Wa,
                          float* __restrict__ gc, float invN) {
    __shared__ float m[64];
    int t = threadIdx.x;
    m[t] = gsum[t] * invN;
    __syncthreads();
    float s = 0.0f;
#pragma unroll 8
    for (int d = 0; d < 64; ++d) s += m[d] * Wa[d * 64 + t];
    gc[t] = tanhf(s);
}

__device__ __forceinline__ float wave_reduce_sum(float v) {
#pragma unroll
    for (int off = 16; off > 0; off >>= 1) v += __shfl_xor(v, off, 32);
    return v;
}

// h = sum_v x[v] * sigmoid(x[v].gc); one wave (32 lanes, 2 cols each) per node
__global__ void attpool_kernel(const float* __restrict__ x, const float* __restrict__ gc,
                               float* __restrict__ h, int N) {
    int lane = threadIdx.x & 31;
    int wave = threadIdx.x >> 5;
    int wavesTotal = gridDim.x * (blockDim.x >> 5);
    int gw = blockIdx.x * (blockDim.x >> 5) + wave;
    float g0 = gc[2 * lane], g1 = gc[2 * lane + 1];
    float a0 = 0.0f, a1 = 0.0f;
    for (long long v = gw; v < N; v += wavesTotal) {
        float x0 = x[v * 64 + 2 * lane];
        float x1 = x[v * 64 + 2 * lane + 1];
        float p = wave_reduce_sum(x0 * g0 + x1 * g1);
        float att = 1.0f / (1.0f + __expf(-p));
        a0 += x0 * att;
        a1 += x1 * att;
    }
    atomicAdd(&h[2 * lane], a0);
    atomicAdd(&h[2 * lane + 1], a1);
}

// ---------------------------------------------------------------------------
// NTN + MLP + score, single block of 256 threads
__global__ void ntn_mlp_score_kernel(const float* __restrict__ hi, const float* __restrict__ hj,
                                     const float* __restrict__ Wt,   // 16*64*64
                                     const float* __restrict__ Wm,   // 16*128
                                     const float* __restrict__ nb,   // 16
                                     const float* __restrict__ w0, const float* __restrict__ b0,
                                     const float* __restrict__ w1, const float* __restrict__ b1,
                                     const float* __restrict__ w2, const float* __restrict__ b2,
                                     const float* __restrict__ w3, const float* __restrict__ b3,
                                     const float* __restrict__ sw, const float* __restrict__ sb,
                                     float* __restrict__ out) {
    __shared__ float shi[64], shj[64], za[32], zb[32], part[256];
    int t = threadIdx.x;
    if (t < 64) { shi[t] = hi[t]; shj[t] = hj[t]; }
    __syncthreads();

    // bilinear: k = t/16, this thread covers rows d = t%16, +16, +32, +48
    int k = t >> 4, p = t & 15;
    float acc = 0.0f;
    for (int d = p; d < 64; d += 16) {
        const float* row = Wt + ((size_t)k * 64 + d) * 64;
        float dot = 0.0f;
#pragma unroll 8
        for (int e = 0; e < 64; ++e) dot += row[e] * shj[e];
        acc += shi[d] * dot;
    }
    part[t] = acc;
    __syncthreads();

    if (t < 16) {
        float s = 0.0f;
        for (int q = 0; q < 16; ++q) s += part[t * 16 + q];
        float lin = 0.0f;
        for (int d = 0; d < 64; ++d)
            lin += Wm[t * 128 + d] * shi[d] + Wm[t * 128 + 64 + d] * shj[d];
        za[t] = tanhf(s + lin + nb[t]);
    }
    __syncthreads();
    if (t < 32) { float s = b0[t]; for (int d = 0; d < 16; ++d) s += za[d] * w0[d * 32 + t]; zb[t] = fmaxf(s, 0.0f); }
    __syncthreads();
    if (t < 16) { float s = b1[t]; for (int d = 0; d < 32; ++d) s += zb[d] * w1[d * 16 + t]; za[t] = fmaxf(s, 0.0f); }
    __syncthreads();
    if (t < 8)  { float s = b2[t]; for (int d = 0; d < 16; ++d) s += za[d] * w2[d * 8 + t];  zb[t] = fmaxf(s, 0.0f); }
    __syncthreads();
    if (t < 4)  { float s = b3[t]; for (int d = 0; d < 8;  ++d) s += zb[d] * w3[d * 4 + t];  za[t] = fmaxf(s, 0.0f); }
    __syncthreads();
    if (t == 0) { float s = sb[0]; for (int d = 0; d < 4;  ++d) s += za[d] * sw[d]; out[0] = s; }
}

// ---------------------------------------------------------------------------
static inline void fill_async(float* p, size_t n, float v, hipStream_t s) {
    int blocks = (int)((n + 255) / 256);
    if (blocks > 4096) blocks = 4096;
    fill_kernel<<<blocks, 256, 0, s>>>(p, n, v);
}

extern "C" void kernel_launch(void* const* d_in, const int* in_sizes, int n_in,
                              void* d_out, int out_size, void* d_ws, size_t ws_size,
                              hipStream_t stream) {
    const float* x_i  = (const float*)d_in[0];
    const int*   ei_i = (const int*)d_in[1];
    const float* x_j  = (const float*)d_in[2];
    const int*   ei_j = (const int*)d_in[3];
    const float* cw[3] = {(const float*)d_in[4], (const float*)d_in[6], (const float*)d_in[8]};
    const float* cb[3] = {(const float*)d_in[5], (const float*)d_in[7], (const float*)d_in[9]};
    const float* att_w  = (const float*)d_in[10];
    const float* ntn_wt = (const float*)d_in[11];
    const float* ntn_wm = (const float*)d_in[12];
    const float* ntn_b  = (const float*)d_in[13];
    const float* mw[4] = {(const float*)d_in[14], (const float*)d_in[16], (const float*)d_in[18], (const float*)d_in[20]};
    const float* mb[4] = {(const float*)d_in[15], (const float*)d_in[17], (const float*)d_in[19], (const float*)d_in[21]};
    const float* sw = (const float*)d_in[22];
    const float* sb = (const float*)d_in[23];

    const int N = in_sizes[0] / D0;
    const int E = in_sizes[1] / 2;

    // workspace layout
    float* ws = (float*)d_ws;
    size_t NB = (size_t)N * DMAX;
    float* xw     = ws;                 // N*128 (shared staging for x@W)
    float* bufA_i = xw + NB;
    float* bufB_i = bufA_i + NB;
    float* bufA_j = bufB_i + NB;
    float* bufB_j = bufA_j + NB;
    float* dinv_i = bufB_j + NB;        // N
    float* dinv_j = dinv_i + N;         // N
    float* sm     = dinv_j + N;         // small scalars
    float* gsum_i = sm;       float* gsum_j = sm + 64;
    float* gc_i   = sm + 128; float* gc_j   = sm + 192;
    float* h_i    = sm + 256; float* h_j    = sm + 320;

    // ---- degrees: deg=1 (self loop) + count(dst), then rsqrt
    fill_async(dinv_i, N, 1.0f, stream);
    fill_async(dinv_j, N, 1.0f, stream);
    count_deg_kernel<<<(E + 255) / 256, 256, 0, stream>>>(ei_i, dinv_i, E);
    count_deg_kernel<<<(E + 255) / 256, 256, 0, stream>>>(ei_j, dinv_j, E);
    rsqrt_kernel<<<(N + 255) / 256, 256, 0, stream>>>(dinv_i, N);
    rsqrt_kernel<<<(N + 255) / 256, 256, 0, stream>>>(dinv_j, N);

    // ---- 3 GCN layers; ReLU of layers 0/1 is fused into the next GEMM's A-load
    const int Kdim[3]   = {64, 128, 128};
    const int Ksh[3]    = {6, 7, 7};
    const int Ddim[3]   = {128, 128, 64};
    const int Dsh[3]    = {7, 7, 6};
    const float* cur_i = x_i;
    const float* cur_j = x_j;
    float* outs_i[3] = {bufA_i, bufB_i, bufA_i};
    float* outs_j[3] = {bufA_j, bufB_j, bufA_j};

    for (int li = 0; li < 3; ++li) {
        const int K = Kdim[li], D = Ddim[li];
        const int kshift = Ksh[li], dshift = Dsh[li], gshift = dshift - 2;
        const int relu_in = (li > 0) ? 1 : 0;        // previous layer's ReLU at load
        dim3 ggrid((N + 15) / 16);
        int gthreads = 32 * (D / 16);                // one wave per 16-col N tile
        size_t nd = (size_t)N << dshift;
        int ibBlocks = (int)((nd + 255) / 256);
        long long aggTotal = (long long)E << gshift;
        int aggBlocks = (int)((aggTotal + 511) / 512);   // ~2 grid-stride iters

        // graph i
        gemm_wmma_f32_lds<<<ggrid, gthreads, 0, stream>>>(cur_i, cw[li], xw, N, K, D, kshift, relu_in);
        init_out_kernel<<<ibBlocks, 256, 0, stream>>>(xw, dinv_i, cb[li], outs_i[li], N, D, dshift);
        aggregate_kernel<<<aggBlocks, 256, 0, stream>>>(xw, outs_i[li], ei_i, dinv_i, E, D, gshift);
        // graph j
        gemm_wmma_f32_lds<<<ggrid, gthreads, 0, stream>>>(cur_j, cw[li], xw, N, K, D, kshift, relu_in);
        init_out_kernel<<<ibBlocks, 256, 0, stream>>>(xw, dinv_j, cb[li], outs_j[li], N, D, dshift);
        aggregate_kernel<<<aggBlocks, 256, 0, stream>>>(xw, outs_j[li], ei_j, dinv_j, E, D, gshift);

        cur_i = outs_i[li];
        cur_j = outs_j[li];
    }

    // ---- attention pooling (final features are 64-wide, un-ReLU'd, in bufA_*)
    fill_async(sm, 384, 0.0f, stream);
    const float invN = 1.0f / (float)N;
    colsum_kernel<<<512, 256, 0, stream>>>(bufA_i, gsum_i, N);
    colsum_kernel<<<512, 256, 0, stream>>>(bufA_j, gsum_j, N);
    gc_kernel<<<1, 64, 0, stream>>>(gsum_i, att_w, gc_i, invN);
    gc_kernel<<<1, 64, 0, stream>>>(gsum_j, att_w, gc_j, invN);
    attpool_kernel<<<256, 256, 0, stream>>>(bufA_i, gc_i, h_i, N);
    attpool_kernel<<<256, 256, 0, stream>>>(bufA_j, gc_j, h_j, N);

    // ---- NTN + MLP + score
    ntn_mlp_score_kernel<<<1, 256, 0, stream>>>(h_i, h_j, ntn_wt, ntn_wm, ntn_b,
                                                mw[0], mb[0], mw[1], mb[1],
                                                mw[2], mb[2], mw[3], mb[3],
                                                sw, sb, (float*)d_out);
}